// HierarchicalRouter_15126874817028
// MI455X (gfx1250) — compile-verified
//
#include <hip/hip_runtime.h>
#include <hip/hip_bf16.h>
#include <math.h>

typedef __attribute__((ext_vector_type(2))) float v2f;
typedef __attribute__((ext_vector_type(8))) float v8f;

#define NUM_EXPERTS 64
#define EGROUP 4
#define NUM_GROUPS 16
#define DDIM 2048
#define NTOK 32768
#define WAVES_PER_BLOCK 8
#define TOK_PER_WAVE 32                                   // two 16-token WMMA tiles per wave
#define TOK_PER_BLOCK (WAVES_PER_BLOCK * TOK_PER_WAVE)    // 256

// d_out layout (floats): final_weights [NTOK*2] | dispatch_mask [NTOK*64] | router_loss [1]
#define FW_CNT (NTOK * 2)
#define DM_CNT (NTOK * NUM_EXPERTS)

// Main kernel: each wave32 computes TWO 16-token tiles of group logits (16x16)
// and local logits (16x16, cols 0-3 valid) with V_WMMA_F32_16X16X4_F32 chains.
// Sharing the Wg/We B-fragments across both tiles halves the weight-stream
// loads per token (loop is 2 A-loads + 2 B-loads : 8 WMMAs per K=8 step).
__global__ __launch_bounds__(256)
void moe_router_main(const float* __restrict__ x,
                     const float* __restrict__ Wg,
                     const float* __restrict__ We,
                     float* __restrict__ out_fw,
                     float* __restrict__ out_dm,
                     float* __restrict__ ws_red) {
    // per-wave staging: 32 tokens x (16 group logits + 4 local logits), padded
    __shared__ float s_log[WAVES_PER_BLOCK][TOK_PER_WAVE][36];

    const int tid  = threadIdx.x;
    const int wave = tid >> 5;
    const int lane = tid & 31;
    const int half = lane >> 4;   // 0: lanes 0-15 (K 0,1), 1: lanes 16-31 (K 2,3)
    const int sub  = lane & 15;   // A: token-in-tile ; B: output column (group/expert)

    const int wtok0 = (blockIdx.x * WAVES_PER_BLOCK + wave) * TOK_PER_WAVE;

    // Contiguous float4 streams per lane (16B aligned: rows are 2048 floats).
    const float4* xrow0 = (const float4*)(x  + (size_t)(wtok0 + sub) * DDIM);
    const float4* xrow1 = (const float4*)(x  + (size_t)(wtok0 + 16 + sub) * DDIM);
    const float4* grow  = (const float4*)(Wg + (size_t)sub * DDIM);
    const float4* erow  = (const float4*)(We + (size_t)(sub & 3) * DDIM); // rows replicated; cols 4-15 ignored

    v8f cg0 = {}, ce0 = {};   // tile 0 accumulators
    v8f cg1 = {}, ce1 = {};   // tile 1 accumulators

    // K super-chunks of 8: one b128 load per stream per lane -> 2 WMMA steps each.
    #pragma unroll 4
    for (int k8 = 0; k8 < DDIM / 8; ++k8) {
        const int idx = k8 * 2 + half;
        float4 av0 = xrow0[idx];
        float4 av1 = xrow1[idx];
        float4 gv  = grow[idx];
        float4 ev  = erow[idx];
        v2f a00 = {av0.x, av0.y}, a01 = {av0.z, av0.w};
        v2f a10 = {av1.x, av1.y}, a11 = {av1.z, av1.w};
        v2f g0  = {gv.x, gv.y},   g1  = {gv.z, gv.w};
        v2f e0  = {ev.x, ev.y},   e1  = {ev.z, ev.w};
        cg0 = __builtin_amdgcn_wmma_f32_16x16x4_f32(false, a00, false, g0, (short)0, cg0, false, false);
        cg0 = __builtin_amdgcn_wmma_f32_16x16x4_f32(false, a01, false, g1, (short)0, cg0, false, false);
        cg1 = __builtin_amdgcn_wmma_f32_16x16x4_f32(false, a10, false, g0, (short)0, cg1, false, false);
        cg1 = __builtin_amdgcn_wmma_f32_16x16x4_f32(false, a11, false, g1, (short)0, cg1, false, false);
        ce0 = __builtin_amdgcn_wmma_f32_16x16x4_f32(false, a00, false, e0, (short)0, ce0, false, false);
        ce0 = __builtin_amdgcn_wmma_f32_16x16x4_f32(false, a01, false, e1, (short)0, ce0, false, false);
        ce1 = __builtin_amdgcn_wmma_f32_16x16x4_f32(false, a10, false, e0, (short)0, ce1, false, false);
        ce1 = __builtin_amdgcn_wmma_f32_16x16x4_f32(false, a11, false, e1, (short)0, ce1, false, false);
    }

    // C/D layout: VGPR i, lanes 0-15 -> (M=i, N=sub); lanes 16-31 -> (M=i+8, N=sub)
    #pragma unroll
    for (int i = 0; i < 8; ++i) {
        const int m = i + half * 8;
        s_log[wave][m][sub]           = cg0[i];
        s_log[wave][m][16 + sub]      = ce0[i];
        s_log[wave][16 + m][sub]      = cg1[i];
        s_log[wave][16 + m][16 + sub] = ce1[i];
    }
    __syncthreads();

    // Every lane post-processes one of the wave's 32 tokens.
    const int t = wtok0 + lane;
    const float* row = &s_log[wave][lane][0];

    // group softmax -> top-1 (ties: lowest index, matching jax top_k)
    float gmax = row[0]; int gidx = 0;
    #pragma unroll
    for (int n = 1; n < NUM_GROUPS; ++n) {
        float v = row[n];
        if (v > gmax) { gmax = v; gidx = n; }
    }
    float zg = 0.0f, gsum = 0.0f;
    #pragma unroll
    for (int n = 0; n < NUM_GROUPS; ++n) {
        float v = row[n];
        zg += v * v;
        gsum += __expf(v - gmax);
    }
    const float gw = 1.0f / gsum;   // top group prob

    // local softmax over 4 experts
    float lv[EGROUP];
    float zl = 0.0f, lmax = -3.0e38f;
    #pragma unroll
    for (int e = 0; e < EGROUP; ++e) {
        lv[e] = row[16 + e];
        zl += lv[e] * lv[e];
        lmax = fmaxf(lmax, lv[e]);
    }
    float p[EGROUP]; float lsum = 0.0f;
    #pragma unroll
    for (int e = 0; e < EGROUP; ++e) { p[e] = __expf(lv[e] - lmax); lsum += p[e]; }
    const float linv = 1.0f / lsum;
    #pragma unroll
    for (int e = 0; e < EGROUP; ++e) p[e] *= linv;

    // top-2 (stable tie-break on lowest index)
    int i1 = 0;
    #pragma unroll
    for (int e = 1; e < EGROUP; ++e) if (p[e] > p[i1]) i1 = e;
    int i2 = (i1 == 0) ? 1 : 0;
    #pragma unroll
    for (int e = 0; e < EGROUP; ++e) if (e != i1 && p[e] > p[i2]) i2 = e;

    const float w1 = p[i1], w2 = p[i2];
    const float dn = w1 + w2 + 1e-7f;
    const float f1 = gw * (w1 / dn);
    const float f2 = gw * (w2 / dn);

    out_fw[(size_t)t * 2 + 0] = f1;
    out_fw[(size_t)t * 2 + 1] = f2;
    const int e1 = gidx * EGROUP + i1;
    const int e2 = gidx * EGROUP + i2;
    out_dm[(size_t)t * NUM_EXPERTS + e1] = f1;
    out_dm[(size_t)t * NUM_EXPERTS + e2] = f2;
    atomicAdd(&ws_red[e1], f1);
    atomicAdd(&ws_red[e2], f2);

    // fold z partials across the full wave
    #pragma unroll
    for (int off = 16; off >= 1; off >>= 1) {
        zg += __shfl_xor(zg, off, 32);
        zl += __shfl_xor(zl, off, 32);
    }
    if (lane == 0) {
        atomicAdd(&ws_red[64], zg);
        atomicAdd(&ws_red[65], zl);
    }
}

// Epilogue: fold expert loads + z sums into the scalar router loss.
__global__ void moe_router_loss(const float* __restrict__ ws_red,
                                float* __restrict__ out_loss) {
    if (threadIdx.x == 0 && blockIdx.x == 0) {
        float total = 0.0f;
        for (int e = 0; e < NUM_EXPERTS; ++e) total += ws_red[e];
        const float target = total / (float)NUM_EXPERTS;
        float lbl = 0.0f;
        for (int e = 0; e < NUM_EXPERTS; ++e) {
            float d = ws_red[e] - target;
            lbl += d * d;
        }
        lbl /= (float)NUM_EXPERTS;
        const float z = ws_red[64] / (float)(NTOK * NUM_GROUPS)
                      + ws_red[65] / (float)(NTOK * EGROUP);
        out_loss[0] = 0.001f * (lbl + z);
    }
}

extern "C" void kernel_launch(void* const* d_in, const int* in_sizes, int n_in,
                              void* d_out, int out_size, void* d_ws, size_t ws_size,
                              hipStream_t stream) {
    const float* x  = (const float*)d_in[0];
    const float* Wg = (const float*)d_in[1];
    const float* We = (const float*)d_in[2];

    float* out  = (float*)d_out;
    float* fw   = out;                   // [NTOK*2]
    float* dm   = out + FW_CNT;          // [NTOK*64]
    float* loss = out + FW_CNT + DM_CNT; // [1]
    float* ws   = (float*)d_ws;          // 66 floats: 64 expert loads + zg + zl

    hipMemsetAsync(ws, 0, 66 * sizeof(float), stream);
    hipMemsetAsync(dm, 0, (size_t)DM_CNT * sizeof(float), stream);

    dim3 grid(NTOK / TOK_PER_BLOCK);     // 128 blocks
    moe_router_main<<<grid, 256, 0, stream>>>(x, Wg, We, fw, dm, ws);
    moe_router_loss<<<1, 32, 0, stream>>>(ws, loss);
}